// GNNEncoder_4741643895614
// MI455X (gfx1250) — compile-verified
//
#include <hip/hip_runtime.h>
#include <hip/hip_bf16.h>

typedef __attribute__((ext_vector_type(2))) float v2f;
typedef __attribute__((ext_vector_type(8))) float v8f;

#define NDIM 128
#define LDS_PITCH 132   // stride 132 ≡ 4 (mod 64 banks) -> conflict-free b64 frag loads

// ---------------------------------------------------------------------------
// Degree / normalization precompute
// ---------------------------------------------------------------------------
__global__ void deg_init_kernel(float* __restrict__ deg, int N) {
    int i = blockIdx.x * blockDim.x + threadIdx.x;
    if (i < N) deg[i] = 1.0f;   // self-loop contributes 1 to every dst degree
}

__global__ void deg_accum_kernel(const long long* __restrict__ dst,
                                 float* __restrict__ deg, int E) {
    int e = blockIdx.x * blockDim.x + threadIdx.x;
    if (e < E) atomicAdd(&deg[(int)dst[e]], 1.0f);
}

__global__ void deg_to_dis_kernel(float* __restrict__ deg, int N) {
    int i = blockIdx.x * blockDim.x + threadIdx.x;
    if (i < N) deg[i] = rsqrtf(deg[i]);   // deg >= 1 always (self-loop)
}

__global__ void norm_kernel(const long long* __restrict__ src,
                            const long long* __restrict__ dst,
                            const float* __restrict__ dis,
                            float* __restrict__ norm, int E) {
    int e = blockIdx.x * blockDim.x + threadIdx.x;
    if (e < E) norm[e] = dis[(int)src[e]] * dis[(int)dst[e]];
}

// ---------------------------------------------------------------------------
// GEMM: out[N,128] = A[N,128] @ W[128,128] (+ bias), via V_WMMA_F32_16X16X4_F32
// 256 threads = 8 waves; W staged TRANSPOSED in LDS so each lane's B fragment
// (W[k][col], W[k+1][col]) is one contiguous ds_load_b64.
// Each wave computes a 16x128 output strip (8 col tiles, A fragment reused).
// ---------------------------------------------------------------------------
__global__ __launch_bounds__(256) void gcn_gemm_wmma(
    const float* __restrict__ A, const float* __restrict__ W,
    const float* __restrict__ bias, float* __restrict__ out, int N)
{
    __shared__ float sWt[NDIM * LDS_PITCH];   // sWt[col * PITCH + k] = W[k][col]

    const int tid = threadIdx.x;
    // Cooperative transposed staging: coalesced float4 global read,
    // 4 scalar LDS stores (one-time cost; inner loop is conflict-free).
    for (int i = tid; i < NDIM * (NDIM / 4); i += 256) {
        int r  = i >> 5;          // k row 0..127
        int c4 = (i & 31) << 2;   // col 0,4,...,124
        const float4 v = *(const float4*)(W + r * NDIM + c4);
        sWt[(c4 + 0) * LDS_PITCH + r] = v.x;
        sWt[(c4 + 1) * LDS_PITCH + r] = v.y;
        sWt[(c4 + 2) * LDS_PITCH + r] = v.z;
        sWt[(c4 + 3) * LDS_PITCH + r] = v.w;
    }
    __syncthreads();

    const int wave  = tid >> 5;
    const int lane  = tid & 31;
    const int half  = lane >> 4;   // 0: lanes 0-15, 1: lanes 16-31
    const int lan16 = lane & 15;

    const int row0 = blockIdx.x * 128 + wave * 16;

    v8f acc[8];
    #pragma unroll
    for (int t = 0; t < 8; ++t)
        #pragma unroll
        for (int r = 0; r < 8; ++r) acc[t][r] = 0.0f;

    // A fragment (ISA 16x4 layout): lane holds M=lan16; VGPR0/1 = K {2h, 2h+1}.
    int rA = row0 + lan16;
    if (rA >= N) rA = N - 1;      // clamp OOB rows (stores are guarded below)
    const float* aRow = A + (long)rA * NDIM;

    #pragma unroll 4
    for (int k0 = 0; k0 < NDIM; k0 += 4) {
        const v2f a = *(const v2f*)(aRow + k0 + half * 2);       // global b64
        #pragma unroll
        for (int t = 0; t < 8; ++t) {
            const int col = t * 16 + lan16;
            const v2f b = *(const v2f*)(&sWt[col * LDS_PITCH + k0 + half * 2]); // ds b64
            acc[t] = __builtin_amdgcn_wmma_f32_16x16x4_f32(
                /*neg_a=*/false, a, /*neg_b=*/false, b,
                /*c_mod=*/(short)0, acc[t],
                /*reuse_a=*/false, /*reuse_b=*/false);
        }
    }

    // D layout: VGPR r holds (M = r + 8*half, N = lan16) for each 16x16 tile.
    #pragma unroll
    for (int t = 0; t < 8; ++t) {
        const int col = t * 16 + lan16;
        const float badd = bias ? bias[col] : 0.0f;
        #pragma unroll
        for (int r = 0; r < 8; ++r) {
            const int rw = row0 + r + 8 * half;
            if (rw < N) out[(long)rw * NDIM + col] = acc[t][r] + badd;
        }
    }
}

// ---------------------------------------------------------------------------
// agg[n,:] = xw[n,:] * dis[n]^2 + bias   (self-loop message + layer bias)
// ---------------------------------------------------------------------------
__global__ void agg_init_kernel(const float* __restrict__ xw,
                                const float* __restrict__ dis,
                                const float* __restrict__ bias,
                                float* __restrict__ agg, int N) {
    int idx = blockIdx.x * blockDim.x + threadIdx.x;   // N * 32 float4 slots
    if (idx >= N * 32) return;
    int node = idx >> 5;
    int c    = (idx & 31) << 2;
    float sn = dis[node];
    sn *= sn;
    float4 v = *(const float4*)(xw + (long)node * NDIM + c);
    float4 b = *(const float4*)(bias + c);
    float4 o;
    o.x = fmaf(v.x, sn, b.x);
    o.y = fmaf(v.y, sn, b.y);
    o.z = fmaf(v.z, sn, b.z);
    o.w = fmaf(v.w, sn, b.w);
    *(float4*)(agg + (long)node * NDIM + c) = o;
}

// ---------------------------------------------------------------------------
// Edge scatter: one wave per edge; lane handles 4 contiguous columns.
// Gather is a coalesced 512B row read (L2-resident), scatter is 4 f32 atomics.
// ---------------------------------------------------------------------------
__global__ __launch_bounds__(256) void edge_scatter_kernel(
    const float* __restrict__ xw,
    const long long* __restrict__ src,
    const long long* __restrict__ dst,
    const float* __restrict__ norm,
    float* __restrict__ agg, int E)
{
    const int lane   = threadIdx.x & 31;
    const int wave   = (blockIdx.x * blockDim.x + threadIdx.x) >> 5;
    const int nwaves = (gridDim.x * blockDim.x) >> 5;
    const int c      = lane << 2;

    for (int e = wave; e < E; e += nwaves) {
        const int   s  = (int)src[e];
        const int   d  = (int)dst[e];
        const float nv = norm[e];
        const float4 v = *(const float4*)(xw + (long)s * NDIM + c);
        float* p = agg + (long)d * NDIM + c;
        atomicAdd(p + 0, v.x * nv);
        atomicAdd(p + 1, v.y * nv);
        atomicAdd(p + 2, v.z * nv);
        atomicAdd(p + 3, v.w * nv);
    }
}

// ---------------------------------------------------------------------------
// ReLU (float4)
// ---------------------------------------------------------------------------
__global__ void relu_kernel(const float* __restrict__ in,
                            float* __restrict__ out, int n4) {
    int idx = blockIdx.x * blockDim.x + threadIdx.x;
    if (idx >= n4) return;
    float4 v = *(const float4*)(in + (long)idx * 4);
    v.x = fmaxf(v.x, 0.0f);
    v.y = fmaxf(v.y, 0.0f);
    v.z = fmaxf(v.z, 0.0f);
    v.w = fmaxf(v.w, 0.0f);
    *(float4*)(out + (long)idx * 4) = v;
}

// ---------------------------------------------------------------------------
// Host-side orchestration
// ---------------------------------------------------------------------------
extern "C" void kernel_launch(void* const* d_in, const int* in_sizes, int n_in,
                              void* d_out, int out_size, void* d_ws, size_t ws_size,
                              hipStream_t stream) {
    const float*     x     = (const float*)d_in[0];
    const long long* eidx  = (const long long*)d_in[1];
    const float*     W_in  = (const float*)d_in[2];
    const float*     b_in  = (const float*)d_in[3];
    const float*     Ws    = (const float*)d_in[4];
    const float*     bs    = (const float*)d_in[5];

    const int N = in_sizes[0] / NDIM;
    const int E = in_sizes[1] / 2;
    const int NUM_LAYERS = in_sizes[4] / (NDIM * NDIM);

    const long long* src = eidx;
    const long long* dst = eidx + E;

    // Workspace carve-up (256B aligned slices)
    char* w = (char*)d_ws;
    auto carve = [&](size_t bytes) {
        char* p = w;
        w += (bytes + 255) & ~(size_t)255;
        return p;
    };
    float* dis  = (float*)carve((size_t)N * sizeof(float));          // deg -> dis in place
    float* norm = (float*)carve((size_t)E * sizeof(float));
    float* h    = (float*)carve((size_t)N * NDIM * sizeof(float));
    float* xw   = (float*)carve((size_t)N * NDIM * sizeof(float));
    float* agg  = (float*)carve((size_t)N * NDIM * sizeof(float));

    const int T = 256;
    const int gN  = (N + T - 1) / T;
    const int gE  = (E + T - 1) / T;
    const int gNv = (N * 32 + T - 1) / T;   // float4 slots over N*128
    const int gGemm = (N + 127) / 128;
    const int gScatter = 2048;              // 16384 waves grid-striding edges

    // 1) degrees -> inverse sqrt -> per-edge norms
    deg_init_kernel<<<gN, T, 0, stream>>>(dis, N);
    deg_accum_kernel<<<gE, T, 0, stream>>>(dst, dis, E);
    deg_to_dis_kernel<<<gN, T, 0, stream>>>(dis, N);
    norm_kernel<<<gE, T, 0, stream>>>(src, dst, dis, norm, E);

    // 2) input projection: h = x @ W_in + b_in
    gcn_gemm_wmma<<<gGemm, T, 0, stream>>>(x, W_in, b_in, h, N);

    // 3) GCN layers
    for (int i = 0; i < NUM_LAYERS; ++i) {
        const float* Wi = Ws + (size_t)i * NDIM * NDIM;
        const float* bi = bs + (size_t)i * NDIM;
        gcn_gemm_wmma<<<gGemm, T, 0, stream>>>(h, Wi, nullptr, xw, N);
        agg_init_kernel<<<gNv, T, 0, stream>>>(xw, dis, bi, agg, N);
        edge_scatter_kernel<<<gScatter, T, 0, stream>>>(xw, src, dst, norm, agg, E);
        float* hout = (i == NUM_LAYERS - 1) ? (float*)d_out : h;
        relu_kernel<<<gNv, T, 0, stream>>>(agg, hout, N * 32);
    }
}